// SliceSumCatOperation_39462159515841
// MI455X (gfx1250) — compile-verified
//
#include <hip/hip_runtime.h>

// ---------------------------------------------------------------------------
// SliceSumCat on MI455X (gfx1250)
//
// out[b, s*64+c] = sum_{r in [start_s, end_s)} x[b, r, c],  8 static slices.
// All slice boundaries are multiples of 25 -> single pass with prefix
// snapshots P25..P200, then out_s = P_end - P_start.
// Memory-bound: ~218 MB total traffic @ 23.3 TB/s  =>  ~9.4 us floor.
// Streaming data is touched exactly once -> nontemporal (TH=NT) loads/stores
// so the 192 MB L2 is not filled with dead lines.
//
// Final combine (8x8 +/-1 matrix applied to 8x64 prefixes per batch) is done
// with V_WMMA_F32_16X16X4_F32 (two chained K=4 steps), operands routed
// through LDS into the ISA-documented wave32 matrix layouts.
// ---------------------------------------------------------------------------

typedef __attribute__((ext_vector_type(2))) float v2f;
typedef __attribute__((ext_vector_type(4))) float v4f;
typedef __attribute__((ext_vector_type(8))) float v8f;

// A-operand (16x4 f32) constant tables for V_WMMA_F32_16X16X4_F32.
// ISA layout: lanes 0-15 hold rows M=0..15; VGPR0 = A[M][K=0] (lanes 0-15)
// and A[M][K=2] (lanes 16-31); VGPR1 = A[M][K=1] / A[M][K=3].
//
// K-step 0 operates on prefixes (P25,P50,P75,P100); K-step 1 on
// (P125,P150,P175,P200).  Row m = slice m coefficients:
//   s0=(0,50)=+P50          s1=(25,100)=P100-P25   s2=(50,150)=P150-P50
//   s3=(0,200)=+P200        s4=(100,200)=P200-P100 s5=(150,200)=P200-P150
//   s6=(0,100)=+P100        s7=(75,175)=P175-P75
__device__ __constant__ float kA0V0[32] = {  // t=0: K0 | K2
    0,-1,0,0,0,0,0,0, 0,0,0,0,0,0,0,0,       // lanes 0-15 : A[m][0] (P25)
    0,0,0,0,0,0,0,-1, 0,0,0,0,0,0,0,0};      // lanes16-31 : A[m][2] (P75)
__device__ __constant__ float kA0V1[32] = {  // t=0: K1 | K3
    1,0,-1,0,0,0,0,0, 0,0,0,0,0,0,0,0,       // A[m][1] (P50)
    0,1,0,0,-1,0,1,0, 0,0,0,0,0,0,0,0};      // A[m][3] (P100)
__device__ __constant__ float kA1V0[32] = {  // t=1: K0 | K2
    0,0,0,0,0,0,0,0, 0,0,0,0,0,0,0,0,        // A[m][0] (P125) all zero
    0,0,0,0,0,0,0,1, 0,0,0,0,0,0,0,0};       // A[m][2] (P175)
__device__ __constant__ float kA1V1[32] = {  // t=1: K1 | K3
    0,0,1,0,0,-1,0,0, 0,0,0,0,0,0,0,0,       // A[m][1] (P150)
    0,0,0,1,1,1,0,0, 0,0,0,0,0,0,0,0};       // A[m][3] (P200)

__global__ __launch_bounds__(256)
void slice_sum_cat_kernel(const float* __restrict__ x,
                          float* __restrict__ out,
                          int nbatch) {
    // LDS: per wave, 8 prefixes x 128 columns (cols 0-63 = batch b0,
    // 64-127 = batch b0+1).  8 waves -> 32 KB (WGP has 320 KB).
    __shared__ __align__(16) float lds[8][8][128];

    const int lane = threadIdx.x & 31;
    const int wave = threadIdx.x >> 5;
    const int gw   = blockIdx.x * 8 + wave;
    const int b0   = gw * 2;                 // this wave handles b0, b0+1
    if (b0 >= nbatch) return;                // wave-uniform exit

    const int bb = lane >> 4;                // 0 -> batch b0, 1 -> batch b0+1
    const int cl = (lane & 15) * 4;          // 4 consecutive columns per lane

    // ---- streaming pass: 200 rows, nontemporal b128 loads, snapshots ----
    const v4f* base = (const v4f*)(x + (size_t)(b0 + bb) * (200 * 64) + cl);
    v4f acc = {0.f, 0.f, 0.f, 0.f};
    v4f snap[8];
#pragma unroll
    for (int seg = 0; seg < 8; ++seg) {
#pragma unroll
        for (int i = 0; i < 25; ++i) {
            const v4f v = __builtin_nontemporal_load(base + (size_t)(seg * 25 + i) * 16);
            acc += v;
        }
        snap[seg] = acc;                     // P25,P50,...,P200
    }

    // ---- snapshots -> LDS (transpose into the WMMA B-operand layout) ----
#pragma unroll
    for (int p = 0; p < 8; ++p)
        *(v4f*)&lds[wave][p][bb * 64 + cl] = snap[p];
    __syncthreads();

    // ---- combine via V_WMMA_F32_16X16X4_F32: D = A1*B1 + (A0*B0 + 0) ----
    const v2f A0 = { kA0V0[lane], kA0V1[lane] };
    const v2f A1 = { kA1V0[lane], kA1V1[lane] };
    const int half = lane >> 4;              // B layout: lanes 0-15 = K0,K1,
    const int nl   = lane & 15;              //           lanes16-31 = K2,K3

#pragma unroll
    for (int c = 0; c < 8; ++c) {            // 8 chunks of 16 columns
        v2f B0, B1;
        // B (4x16 f32): VGPR0 = K0 (lanes 0-15) | K2 (lanes 16-31),
        //               VGPR1 = K1 | K3.  Prefix index p = 4*t + K for step t.
        B0.x = lds[wave][2 * half + 0][c * 16 + nl];   // P25  / P75
        B0.y = lds[wave][2 * half + 1][c * 16 + nl];   // P50  / P100
        B1.x = lds[wave][2 * half + 4][c * 16 + nl];   // P125 / P175
        B1.y = lds[wave][2 * half + 5][c * 16 + nl];   // P150 / P200

        v8f d = {0.f, 0.f, 0.f, 0.f, 0.f, 0.f, 0.f, 0.f};
        d = __builtin_amdgcn_wmma_f32_16x16x4_f32(false, A0, false, B0,
                                                  (short)0, d, false, false);
        d = __builtin_amdgcn_wmma_f32_16x16x4_f32(false, A1, false, B1,
                                                  (short)0, d, false, false);

        // D layout: VGPR v = slice v (lanes 0-15, N = lane).  Rows 8-15 of
        // the A matrix are zero, so lanes 16-31 hold zeros -> only lanes
        // 0-15 store.
        const int   ob = b0 + (c >> 2);
        float*      op = out + (size_t)ob * 512 + (c & 3) * 16 + nl;
        if (lane < 16) {
#pragma unroll
            for (int s = 0; s < 8; ++s)
                __builtin_nontemporal_store(d[s], op + s * 64);
        }
    }
}

extern "C" void kernel_launch(void* const* d_in, const int* in_sizes, int n_in,
                              void* d_out, int out_size, void* d_ws, size_t ws_size,
                              hipStream_t stream) {
    const float* x   = (const float*)d_in[0];
    float*       out = (float*)d_out;
    const int nbatch = in_sizes[0] / (200 * 64);   // 4096
    const int waves  = (nbatch + 1) / 2;           // 2 batches per wave
    const int blocks = (waves + 7) / 8;            // 8 waves per block
    slice_sum_cat_kernel<<<blocks, 256, 0, stream>>>(x, out, nbatch);
}